// OSI_55791625175663
// MI455X (gfx1250) — compile-verified
//
#include <hip/hip_runtime.h>

#define NTOT 409600
#define BB 4
#define HH 320
#define WW 320
#define HWP 102400

typedef __attribute__((ext_vector_type(2))) float v2f;
typedef __attribute__((ext_vector_type(8))) float v8f;

// ---------------- constant quadrature / reduction tables -------------------
// K-slots 0..8  : edge quadrature points (i=k/3, j=k%3)
// K-slots 9..11 : node quadrature (XI pre-scaled by sqrt(2), XJ=0)
// A rows: 0:XI 1:XJ 2:XI^2+XJ^2-1 3:XI^2-XJ^2 4:XI*XJ 5:XN(node) 6:2XN^2-1(node)
struct QTab {
    float XI[12];
    float XJ[12];
    float W[12];
    float A[16][12];
};

constexpr QTab makeTab() {
    QTab t{};
    const double x[3]  = {-1.2247448713915890491, 0.0, 1.2247448713915890491};
    const double wn[3] = {1.0 / 6.0, 2.0 / 3.0, 1.0 / 6.0};
    const double S2 = 1.4142135623730951455;
    for (int k = 0; k < 12; k++) {
        t.XI[k] = 0.f; t.XJ[k] = 0.f; t.W[k] = 0.f;
        for (int r = 0; r < 16; r++) t.A[r][k] = 0.f;
    }
    for (int k = 0; k < 9; k++) {
        int i = k / 3, j = k % 3;
        double xi = x[i], xj = x[j];
        t.XI[k] = (float)xi;
        t.XJ[k] = (float)xj;
        t.W[k]  = (float)(wn[i] * wn[j]);
        t.A[0][k] = (float)xi;
        t.A[1][k] = (float)xj;
        t.A[2][k] = (float)(xi * xi + xj * xj - 1.0);
        t.A[3][k] = (float)(xi * xi - xj * xj);
        t.A[4][k] = (float)(xi * xj);
    }
    for (int j = 0; j < 3; j++) {
        int k = 9 + j;
        t.XI[k] = (float)(x[j] * S2);
        t.XJ[k] = 0.f;
        t.W[k]  = (float)wn[j];
        t.A[5][k] = (float)x[j];
        t.A[6][k] = (float)(2.0 * x[j] * x[j] - 1.0);
    }
    return t;
}

__constant__ QTab gT = makeTab();

// ws float-plane layout (N = 409600 each):
// 0 mu | 1 sigma | 2 rou0 | 3 rou1 | 4..7 v0..v3
// 8 gmu_self | 9 gsig_self | 10 grou0 | 11 grou1
// 12 dmu2_down | 13 dmu2_right | 14 dsig2_down | 15 dsig2_right
// [16*N] : wmax scalar

// ------------------------- max reduction over weights ----------------------
__global__ __launch_bounds__(256) void OSI_redmax(const float* __restrict__ w,
                                                  int n, float* __restrict__ out) {
    __shared__ float sm[256];
    float m = 0.f;
    for (int i = blockIdx.x * blockDim.x + threadIdx.x; i < n; i += gridDim.x * blockDim.x)
        m = fmaxf(m, w[i]);
    sm[threadIdx.x] = m;
    __syncthreads();
    for (int s = 128; s > 0; s >>= 1) {
        if (threadIdx.x < s) sm[threadIdx.x] = fmaxf(sm[threadIdx.x], sm[threadIdx.x + s]);
        __syncthreads();
    }
    if (threadIdx.x == 0) atomicMax((int*)out, __float_as_int(sm[0]));  // weights >= 0
}

// ------------------------------- init --------------------------------------
__global__ __launch_bounds__(256) void OSI_init(const float* __restrict__ unary,
                                                const float* __restrict__ wgt,
                                                float* __restrict__ ws) {
    int idx = blockIdx.x * 256 + threadIdx.x;
    if (idx >= NTOT) return;
    const int N = NTOT;
    float denom = ws[16 * N] * 1.01f + 1.0f;
    ws[0 * N + idx] = unary[idx];
    ws[1 * N + idx] = 1.0f;
    ws[2 * N + idx] = wgt[2 * idx + 0] / denom;
    ws[3 * N + idx] = wgt[2 * idx + 1] / denom;
    ws[4 * N + idx] = 0.f;
    ws[5 * N + idx] = 0.f;
    ws[6 * N + idx] = 0.f;
    ws[7 * N + idx] = 0.f;
}

// -------------------- gradient pass: quadrature via WMMA -------------------
// one wave = 16 pixels; lane L and L+16 cooperate on pixel (L&15),
// splitting the 12 K-columns per the f32 WMMA A/B VGPR layouts.
// D = A_const(16x4-chunks) x B_data(4x16 pixels): row m of D for pixel p lands
// in VGPR m, lane p -> no cross-lane shuffles needed anywhere.
__global__ __launch_bounds__(256) void OSI_grad(const float* __restrict__ unary,
                                                const float* __restrict__ wgt,
                                                float* __restrict__ ws) {
    const int N = NTOT;
    const float* mu = ws + 0 * N;
    const float* sg = ws + 1 * N;
    const float* ro = ws + 2 * N;  // rou0 plane 2, rou1 plane 3

    int tid  = blockIdx.x * 256 + threadIdx.x;
    int wave = tid >> 5;
    int lane = threadIdx.x & 31;
    int rowi = lane & 15;
    int h    = lane >> 4;
    int idx  = wave * 16 + rowi;

    int b = idx / HWP;
    int rem = idx - b * HWP;
    int y = rem / WW;
    int x = rem - y * WW;
    int yD = (y + 1 == HH) ? 0 : y + 1;
    int xR = (x + 1 == WW) ? 0 : x + 1;
    int iD = b * HWP + yD * WW + x;
    int iR = b * HWP + y * WW + xR;

    float u  = unary[idx];
    float m  = mu[idx];
    float s  = sg[idx];
    float r0 = ro[idx];
    float r1 = ro[N + idx];
    float w0 = wgt[2 * idx + 0];
    float w1 = wgt[2 * idx + 1];
    float mD = mu[iD], sD = sg[iD];
    float mR = mu[iR], sR = sg[iR];

    const float SQ2  = 1.41421356237f;
    const float NDCc = 0.9189385332f;

    float lns = __logf(s);
    // edge 0 (down)
    float q0 = 1.f - r0 * r0;
    float rtq0 = sqrtf(q0);
    float sp0 = sqrtf(1.f + r0) * 0.5f, rn0 = sqrtf(1.f - r0) * 0.5f;
    float a0 = sp0 + rn0, bE0 = sp0 - rn0;
    float P0 = SQ2 * (s * a0 - sD * bE0);
    float Q0 = SQ2 * (s * bE0 - sD * a0);
    float R0 = m - mD;
    float L0 = 2.f * NDCc + lns + __logf(sD) + 0.5f * __logf(q0);
    // edge 1 (right)
    float q1 = 1.f - r1 * r1;
    float rtq1 = sqrtf(q1);
    float sp1 = sqrtf(1.f + r1) * 0.5f, rn1 = sqrtf(1.f - r1) * 0.5f;
    float a1 = sp1 + rn1, bE1 = sp1 - rn1;
    float P1 = SQ2 * (s * a1 - sR * bE1);
    float Q1 = SQ2 * (s * bE1 - sR * a1);
    float R1 = m - mR;
    float L1 = 2.f * NDCc + lns + __logf(sR) + 0.5f * __logf(q1);
    // node (uniform slot form: d = P*xi + Q*xj + R with P=s, Q=0, R=mu-u,
    // weight 1, base term -3*(NDC+ln(sigma)); XI slots 9..11 carry sqrt(2)*XN)
    float tb = m - u;
    float nconst = -3.f * (NDCc + lns);

    v8f acc0 = {};  // edge0 rows 0..4, node rows 5..6
    v8f acc1 = {};  // edge1 rows 0..4
#pragma unroll
    for (int c = 0; c < 3; c++) {
        int k0 = 4 * c + 2 * h;
        int k1 = k0 + 1;
        // unconditional table loads -> no divergent control flow
        float xi0 = gT.XI[k0], xj0 = gT.XJ[k0], wk0 = gT.W[k0];
        float xi1 = gT.XI[k1], xj1 = gT.XJ[k1], wk1 = gT.W[k1];
        v2f Ac = {gT.A[rowi][k0], gT.A[rowi][k1]};

        float b00, b01, b10, b11;
        if (c < 2) {
            // statically edge-only slots (k <= 7)
            float d00 = P0 * xi0 + Q0 * xj0 + R0;
            float d01 = P0 * xi1 + Q0 * xj1 + R0;
            float d10 = P1 * xi0 + Q1 * xj0 + R1;
            float d11 = P1 * xi1 + Q1 * xj1 + R1;
            b00 = (L0 - w0 * d00 * d00) * wk0;
            b01 = (L0 - w0 * d01 * d01) * wk1;
            b10 = (L1 - w1 * d10 * d10) * wk0;
            b11 = (L1 - w1 * d11 * d11) * wk1;
        } else {
            // mixed slots: lo half {8 edge, 9 node}; hi half {10 node, 11 node}
            // branchless: select slot params (P,Q,R,w,L) via v_cndmask
            bool n0 = (k0 >= 9);  // slot k0 is a node slot (true iff h==1)
            float Pa = n0 ? s : P0;
            float Qa = n0 ? 0.f : Q0;
            float Ra = n0 ? tb : R0;
            float wa = n0 ? 1.f : w0;
            float La = n0 ? nconst : L0;
            float d00 = Pa * xi0 + Qa * xj0 + Ra;
            b00 = (La - wa * d00 * d00) * wk0;
            // slot k1 (9 or 11) is always a node slot
            float d01 = s * xi1 + tb;
            b01 = (nconst - d01 * d01) * wk1;
            // chain 1: edge value for k0<9, zero padding for node slots
            float d10 = P1 * xi0 + Q1 * xj0 + R1;
            float e10 = (L1 - w1 * d10 * d10) * wk0;
            b10 = n0 ? 0.f : e10;
            b11 = 0.f;
        }
        v2f B0c = {b00, b01};
        v2f B1c = {b10, b11};
        acc0 = __builtin_amdgcn_wmma_f32_16x16x4_f32(false, Ac, false, B0c, (short)0,
                                                     acc0, false, false);
        acc1 = __builtin_amdgcn_wmma_f32_16x16x4_f32(false, Ac, false, B1c, (short)0,
                                                     acc1, false, false);
    }

    if (h == 0) {  // rows 0..7 of D live in lanes 0..15, column = pixel = lane
        float e00 = acc0[0], e01 = acc0[1], e02 = acc0[2], e03 = acc0[3], e04 = acc0[4];
        float n1 = acc0[5], n2 = acc0[6];
        float e10 = acc1[0], e11 = acc1[1], e12 = acc1[2], e13 = acc1[3], e14 = acc1[4];

        float inv_s = 1.f / s;
        float c0p = a0 - r0 * bE0, c0m = bE0 - r0 * a0;
        float c1p = a1 - r1 * bE1, c1m = bE1 - r1 * a1;

        float dmu = n1 * SQ2 * inv_s;
        dmu += (c0p * e00 + c0m * e01) * SQ2 * inv_s / q0;
        dmu += (c1p * e10 + c1m * e11) * SQ2 * inv_s / q1;
        float d2_0 = (c0m * e00 + c0p * e01) * SQ2 / (sD * q0);
        float d2_1 = (c1m * e10 + c1p * e11) * SQ2 / (sR * q1);

        float dsg = n2 * inv_s;
        dsg += (e02 + e03 / rtq0) * inv_s;
        dsg += (e12 + e13 / rtq1) * inv_s;
        float s2_0 = (e02 - e03 / rtq0) / sD;
        float s2_1 = (e12 - e13 / rtq1) / sR;

        float dr0 = (2.f * e04 - r0 * e02) / q0;
        float dr1 = (2.f * e14 - r1 * e12) / q1;

        ws[8 * N + idx]  = dmu;
        ws[9 * N + idx]  = dsg;
        ws[10 * N + idx] = dr0;
        ws[11 * N + idx] = dr1;
        ws[12 * N + idx] = d2_0;
        ws[13 * N + idx] = d2_1;
        ws[14 * N + idx] = s2_0;
        ws[15 * N + idx] = s2_1;
    }
}

// --------------- update pass: gather neighbor grads, momentum --------------
__global__ __launch_bounds__(256) void OSI_update(float* __restrict__ ws) {
    const int N = NTOT;
    int idx = blockIdx.x * 256 + threadIdx.x;
    if (idx >= N) return;
    int b = idx / HWP;
    int rem = idx - b * HWP;
    int y = rem / WW;
    int x = rem - y * WW;
    int yU = (y == 0) ? HH - 1 : y - 1;
    int xL = (x == 0) ? WW - 1 : x - 1;
    int iU = b * HWP + yU * WW + x;
    int iL = b * HWP + y * WW + xL;

    float dmu = ws[8 * N + idx] + ws[12 * N + iU] + ws[13 * N + iL];
    float dsg = ws[9 * N + idx] + ws[14 * N + iU] + ws[15 * N + iL];
    float dr0 = ws[10 * N + idx];
    float dr1 = ws[11 * N + idx];

    float v0 = 0.7f * ws[4 * N + idx] + 0.01f * dmu;
    float v1 = 0.7f * ws[5 * N + idx] + 0.01f * dsg;
    float v2 = 0.7f * ws[6 * N + idx] + 0.01f * dr0;
    float v3 = 0.7f * ws[7 * N + idx] + 0.01f * dr1;
    ws[4 * N + idx] = v0;
    ws[5 * N + idx] = v1;
    ws[6 * N + idx] = v2;
    ws[7 * N + idx] = v3;

    ws[0 * N + idx] = fminf(fmaxf(ws[0 * N + idx] + v0, 0.0f), 63.0f);
    ws[1 * N + idx] = fminf(fmaxf(ws[1 * N + idx] + v1, 0.001f), 50.0f);
    ws[2 * N + idx] = fminf(fmaxf(ws[2 * N + idx] + v2, -0.99f), 0.99f);
    ws[3 * N + idx] = fminf(fmaxf(ws[3 * N + idx] + v3, -0.99f), 0.99f);
}

__global__ __launch_bounds__(256) void OSI_copyout(const float* __restrict__ ws,
                                                   float* __restrict__ out) {
    int idx = blockIdx.x * 256 + threadIdx.x;
    if (idx < NTOT) out[idx] = ws[idx];
}

// ---------------------------------------------------------------------------
extern "C" void kernel_launch(void* const* d_in, const int* in_sizes, int n_in,
                              void* d_out, int out_size, void* d_ws, size_t ws_size,
                              hipStream_t stream) {
    const float* unary = (const float*)d_in[0];  // [4,320,320]
    const float* wgt   = (const float*)d_in[1];  // [4,320,320,2]
    float* ws = (float*)d_ws;
    float* wmax = ws + 16 * (size_t)NTOT;

    hipMemsetAsync(wmax, 0, sizeof(float), stream);
    OSI_redmax<<<512, 256, 0, stream>>>(wgt, 2 * NTOT, wmax);
    OSI_init<<<NTOT / 256, 256, 0, stream>>>(unary, wgt, ws);

    for (int it = 0; it < 100; ++it) {
        OSI_grad<<<(NTOT / 16) * 32 / 256, 256, 0, stream>>>(unary, wgt, ws);
        OSI_update<<<NTOT / 256, 256, 0, stream>>>(ws);
    }
    OSI_copyout<<<NTOT / 256, 256, 0, stream>>>(ws, (float*)d_out);
}